// SpatialEncoder_79096117723677
// MI455X (gfx1250) — compile-verified
//
#include <hip/hip_runtime.h>
#include <hip/hip_bf16.h>
#include <math.h>

// ---------------------------------------------------------------------------
// 2-layer GATConv (PyG semantics) for MI455X / gfx1250.
// fp32 throughout (edge-scatter bound, not matrix bound); GEMMs use
// V_WMMA_F32_16X16X4_F32. Segment ops via L2-resident float atomics.
// ---------------------------------------------------------------------------

typedef __attribute__((ext_vector_type(2))) float v2f;
typedef __attribute__((ext_vector_type(8))) float v8f;

#define NEG_SLOPE 0.2f

// ---------------- utility kernels ----------------

__global__ void fill_kernel(float* __restrict__ p, float v, int n) {
    int i = blockIdx.x * blockDim.x + threadIdx.x;
    if (i < n) p[i] = v;
}

__global__ void bias_relu_kernel(float* __restrict__ h, const float* __restrict__ b,
                                 int n, int c) {
    int i = blockIdx.x * blockDim.x + threadIdx.x;
    if (i >= n * c) return;
    float v = h[i] + b[i % c];
    h[i] = v > 0.f ? v : 0.f;
}

__global__ void bias_add_kernel(float* __restrict__ h, const float* __restrict__ b,
                                int n, int c) {
    int i = blockIdx.x * blockDim.x + threadIdx.x;
    if (i >= n * c) return;
    h[i] += b[i % c];
}

// ---------------- WMMA fp32 GEMM: Y[M,NOUT] = X[M,K] @ W[K,NOUT] ----------------
// One 16-row tile per block; blockDim = 128 (4 waves), each wave owns one
// 16-wide column tile of NOUT=64. K-loop of 16x16x4 fp32 WMMA steps.
// A 16x4 frag: lane L -> row = L&15, K-pair = (L>>4)*2  (ISA 7.12.2 layout)
// B 4x16 frag: lane L -> col = L&15, K-pair = (L>>4)*2
// C/D 16x16:   VGPR v, lane L -> row = v + 8*(L>>4), col = L&15
template<int K, int NOUT>
__global__ void gemm_wmma_f32(const float* __restrict__ X, const float* __restrict__ W,
                              float* __restrict__ Y) {
    const int lane = threadIdx.x & 31;
    const int wave = threadIdx.x >> 5;          // 0..3 -> column tile
    const int lm   = lane & 15;
    const int lh   = lane >> 4;                 // 0 or 1
    const int row0 = blockIdx.x * 16;
    const int col0 = wave * 16;
    const int arow = row0 + lm;

    v8f acc = {};
    for (int k0 = 0; k0 < K; k0 += 4) {
        const int kb = k0 + lh * 2;
        v2f a, b;
        a.x = X[(long long)arow * K + kb];
        a.y = X[(long long)arow * K + kb + 1];
        b.x = W[kb * NOUT + col0 + lm];
        b.y = W[(kb + 1) * NOUT + col0 + lm];
        acc = __builtin_amdgcn_wmma_f32_16x16x4_f32(
            /*neg_a=*/false, a, /*neg_b=*/false, b,
            /*c_mod=*/(short)0, acc, /*reuse_a=*/false, /*reuse_b=*/false);
    }
#pragma unroll
    for (int v = 0; v < 8; ++v) {
        const int r = row0 + v + 8 * lh;
        Y[(long long)r * NOUT + col0 + lm] = acc[v];
    }
}

// ---------------- attention logits: als/ald[i,h] = dot(H[i,h,:], a_s/a_d[h,:]) ----
template<int H, int C>
__global__ void att_logits_kernel(const float* __restrict__ Hm,
                                  const float* __restrict__ a_s,
                                  const float* __restrict__ a_d,
                                  float* __restrict__ als, float* __restrict__ ald,
                                  int n) {
    int i = blockIdx.x * blockDim.x + threadIdx.x;
    if (i >= n) return;
    const float* row = Hm + (long long)i * H * C;
#pragma unroll
    for (int h = 0; h < H; ++h) {
        float s = 0.f, d = 0.f;
        for (int c = 0; c < C; ++c) {
            float v = row[h * C + c];
            s += v * a_s[h * C + c];
            d += v * a_d[h * C + c];
        }
        als[i * H + h] = s;
        ald[i * H + h] = d;
    }
}

// ---------------- float atomic max via int/uint ordering trick ----------------
__device__ __forceinline__ void atomicMaxFloat(float* addr, float val) {
    if (val >= 0.f) atomicMax((int*)addr, __float_as_int(val));
    else            atomicMin((unsigned int*)addr, __float_as_uint(val));
}

__device__ __forceinline__ float leaky(float v) {
    return v > 0.f ? v : NEG_SLOPE * v;
}

// ---------------- edge pass 1: segment max of leaky(als[src]+ald[dst]) --------
template<int H>
__global__ void edge_max_kernel(const int* __restrict__ src, const int* __restrict__ dst,
                                int E_, int NE,
                                const float* __restrict__ als, const float* __restrict__ ald,
                                float* __restrict__ emax) {
    int e = blockIdx.x * blockDim.x + threadIdx.x;
    if (e >= NE) return;
    int s = (e < E_) ? src[e] : (e - E_);   // self loops appended
    int d = (e < E_) ? dst[e] : (e - E_);
#pragma unroll
    for (int h = 0; h < H; ++h) {
        float v = leaky(als[s * H + h] + ald[d * H + h]);
        atomicMaxFloat(&emax[d * H + h], v);
    }
}

// ---------------- edge pass 2: ex = exp(e - emax[dst]); den[dst] += ex --------
template<int H>
__global__ void edge_expsum_kernel(const int* __restrict__ src, const int* __restrict__ dst,
                                   int E_, int NE,
                                   const float* __restrict__ als, const float* __restrict__ ald,
                                   const float* __restrict__ emax,
                                   float* __restrict__ ex, float* __restrict__ den) {
    int e = blockIdx.x * blockDim.x + threadIdx.x;
    if (e >= NE) return;
    int s = (e < E_) ? src[e] : (e - E_);
    int d = (e < E_) ? dst[e] : (e - E_);
#pragma unroll
    for (int h = 0; h < H; ++h) {
        float v  = leaky(als[s * H + h] + ald[d * H + h]);
        float xv = expf(v - emax[d * H + h]);
        ex[(long long)e * H + h] = xv;
        atomicAdd(&den[d * H + h], xv);
    }
}

// ---------------- edge pass 3: agg[dst] += h[src] * ex/(den[dst]+eps) ---------
// One thread per (edge, 4 channels): float4 gather + 4 float atomics (L2-resident).
template<int H, int C>
__global__ void edge_agg_kernel(const int* __restrict__ src, const int* __restrict__ dst,
                                int E_, int NE,
                                const float* __restrict__ Hm,
                                const float* __restrict__ ex, const float* __restrict__ den,
                                float* __restrict__ agg) {
    const int HC = H * C;
    const int Q  = HC / 4;                      // float4 groups per edge
    long long idx = (long long)blockIdx.x * blockDim.x + threadIdx.x;
    if (idx >= (long long)NE * Q) return;
    int e = (int)(idx / Q);
    int q = (int)(idx % Q) * 4;                 // channel base in [0, HC)
    int s = (e < E_) ? src[e] : (e - E_);
    int d = (e < E_) ? dst[e] : (e - E_);
    int h = q / C;
    float coef = ex[(long long)e * H + h] / (den[d * H + h] + 1e-16f);
    const float4 hv = *(const float4*)(Hm + (long long)s * HC + q);
    float* out = agg + (long long)d * HC + q;
    atomicAdd(out + 0, hv.x * coef);
    atomicAdd(out + 1, hv.y * coef);
    atomicAdd(out + 2, hv.z * coef);
    atomicAdd(out + 3, hv.w * coef);
}

// ---------------------------------------------------------------------------

extern "C" void kernel_launch(void* const* d_in, const int* in_sizes, int n_in,
                              void* d_out, int out_size, void* d_ws, size_t ws_size,
                              hipStream_t stream) {
    (void)n_in; (void)out_size; (void)ws_size;

    const float* x      = (const float*)d_in[0];
    const int*   eidx   = (const int*)d_in[1];      // [2, E] int32 (JAX x64 off)
    const float* W1     = (const float*)d_in[2];
    const float* a_src1 = (const float*)d_in[3];
    const float* a_dst1 = (const float*)d_in[4];
    const float* b1     = (const float*)d_in[5];
    const float* W2     = (const float*)d_in[6];
    const float* a_src2 = (const float*)d_in[7];
    const float* a_dst2 = (const float*)d_in[8];
    const float* b2     = (const float*)d_in[9];
    float* outp = (float*)d_out;

    const int IN = 128, HID = 32, HEADS = 2, OUT = 64, HC = HEADS * HID; // HC==64
    const int N  = in_sizes[0] / IN;            // 50000 (== 3125 * 16, no tail)
    const int E  = in_sizes[1] / 2;             // 800000
    const int NE = E + N;                       // with self loops
    const int* srcp = eidx;
    const int* dstp = eidx + E;

    // ---- workspace layout (floats) ----
    float* w = (float*)d_ws;
    size_t o = 0;
    float* h1_pre = w + o; o += (size_t)N * HC;   // x @ W1
    float* agg1   = w + o; o += (size_t)N * HC;   // layer-1 aggregate, then relu'd h1
    float* h2_pre = w + o; o += (size_t)N * OUT;  // h1 @ W2
    float* als1 = w + o; o += (size_t)N * HEADS;
    float* ald1 = w + o; o += (size_t)N * HEADS;
    float* emax1 = w + o; o += (size_t)N * HEADS;
    float* den1  = w + o; o += (size_t)N * HEADS;
    float* als2 = w + o; o += (size_t)N;
    float* ald2 = w + o; o += (size_t)N;
    float* emax2 = w + o; o += (size_t)N;
    float* den2  = w + o; o += (size_t)N;
    float* ex1 = w + o; o += (size_t)NE * HEADS;
    float* ex2 = w + o; o += (size_t)NE;

    const int B = 256;
    auto blocks = [](long long n, int b) { return (unsigned)((n + b - 1) / b); };
    const float NEG_INF = -__builtin_huge_valf();

    // ================= Layer 1 (2 heads x 32 ch, concat) =================
    fill_kernel<<<blocks((long long)N * HC, B), B, 0, stream>>>(agg1, 0.f, N * HC);
    fill_kernel<<<blocks(N * HEADS, B), B, 0, stream>>>(emax1, NEG_INF, N * HEADS);
    fill_kernel<<<blocks(N * HEADS, B), B, 0, stream>>>(den1, 0.f, N * HEADS);

    gemm_wmma_f32<128, 64><<<N / 16, 128, 0, stream>>>(x, W1, h1_pre);

    att_logits_kernel<2, 32><<<blocks(N, B), B, 0, stream>>>(h1_pre, a_src1, a_dst1,
                                                             als1, ald1, N);
    edge_max_kernel<2><<<blocks(NE, B), B, 0, stream>>>(srcp, dstp, E, NE, als1, ald1, emax1);
    edge_expsum_kernel<2><<<blocks(NE, B), B, 0, stream>>>(srcp, dstp, E, NE, als1, ald1,
                                                           emax1, ex1, den1);
    edge_agg_kernel<2, 32><<<blocks((long long)NE * (HC / 4), B), B, 0, stream>>>(
        srcp, dstp, E, NE, h1_pre, ex1, den1, agg1);

    bias_relu_kernel<<<blocks((long long)N * HC, B), B, 0, stream>>>(agg1, b1, N, HC);

    // ================= Layer 2 (1 head x 64 ch, mean==identity) ==========
    fill_kernel<<<blocks((long long)N * OUT, B), B, 0, stream>>>(outp, 0.f, N * OUT);
    fill_kernel<<<blocks(N, B), B, 0, stream>>>(emax2, NEG_INF, N);
    fill_kernel<<<blocks(N, B), B, 0, stream>>>(den2, 0.f, N);

    gemm_wmma_f32<64, 64><<<N / 16, 128, 0, stream>>>(agg1, W2, h2_pre);

    att_logits_kernel<1, 64><<<blocks(N, B), B, 0, stream>>>(h2_pre, a_src2, a_dst2,
                                                             als2, ald2, N);
    edge_max_kernel<1><<<blocks(NE, B), B, 0, stream>>>(srcp, dstp, E, NE, als2, ald2, emax2);
    edge_expsum_kernel<1><<<blocks(NE, B), B, 0, stream>>>(srcp, dstp, E, NE, als2, ald2,
                                                           emax2, ex2, den2);
    edge_agg_kernel<1, 64><<<blocks((long long)NE * (OUT / 4), B), B, 0, stream>>>(
        srcp, dstp, E, NE, h2_pre, ex2, den2, outp);

    bias_add_kernel<<<blocks((long long)N * OUT, B), B, 0, stream>>>(outp, b2, N, OUT);
}